// TotalLoss_66709432041668
// MI455X (gfx1250) — compile-verified
//
#include <hip/hip_runtime.h>

// ---------------------------------------------------------------------------
// TotalLoss for MI455X (gfx1250, wave32).
// Streaming fp32 reduction: ~168 MB read once -> ~7.2 us at 23.3 TB/s HBM.
// Uses V_WMMA_F32_16X16X4_F32 (exact fp32) as the intra-wave reduction tree.
// ---------------------------------------------------------------------------

typedef float v2f __attribute__((ext_vector_type(2)));
typedef float v4f __attribute__((ext_vector_type(4)));
typedef float v8f __attribute__((ext_vector_type(8)));

#define NT(p) __builtin_nontemporal_load(p)

// Exact full-wave32 sum using the fp32 WMMA.
// A layout (ISA 7.12.2, 32-bit A 16x4): lane m (0-15): vgpr0=A[m][0], vgpr1=A[m][1];
// lane 16+m: vgpr0=A[m][2], vgpr1=A[m][3].  Put per-lane value in vgpr0, 0 in vgpr1.
// B = ones(4x16)  =>  D[m][n] = v_m + v_{m+16}  (row-constant).
// D vgpr j: lanes 0-15 hold R_j, lanes 16-31 hold R_{j+8}.  Sum 8 vgprs per lane,
// then combine lane-halves with one shfl_xor(16).  All adds are fp32-exact order.
__device__ __forceinline__ float wave_sum32(float v) {
  v2f a; a[0] = v;    a[1] = 0.0f;
  v2f b; b[0] = 1.0f; b[1] = 1.0f;
  v8f c = {};
  v8f d = __builtin_amdgcn_wmma_f32_16x16x4_f32(
      /*neg_a=*/false, a, /*neg_b=*/false, b,
      /*c_mod=*/(short)0, c, /*reuse_a=*/false, /*reuse_b=*/false);
  float h = ((d[0] + d[1]) + (d[2] + d[3])) + ((d[4] + d[5]) + (d[6] + d[7]));
  h += __shfl_xor(h, 16, 32);
  return h;
}

__global__ __launch_bounds__(256)
void tl_partial(const float* __restrict__ resnet,
                const float* __restrict__ gru,
                const float* __restrict__ avgp,
                const float* __restrict__ yv,
                const float* __restrict__ avgh,
                float* __restrict__ ws,
                int n, int nseg) {
  const int tid    = blockIdx.x * blockDim.x + threadIdx.x;
  const int stride = gridDim.x * blockDim.x;

  float s1a = 0.0f, s1b = 0.0f, s2 = 0.0f;

  // ---- L1 terms: |resnet - y| and |avg_pred - avg_hr|, b128 NT streams ----
  const int nv = n >> 2;
  const v4f* r4 = (const v4f*)resnet;
  const v4f* y4 = (const v4f*)yv;
  const v4f* p4 = (const v4f*)avgp;
  const v4f* h4 = (const v4f*)avgh;
  for (int i = tid; i < nv; i += stride) {
    v4f ra = NT(r4 + i), yb = NT(y4 + i);
    v4f pa = NT(p4 + i), hb = NT(h4 + i);
    s1a += (fabsf(ra[0] - yb[0]) + fabsf(ra[1] - yb[1])) +
           (fabsf(ra[2] - yb[2]) + fabsf(ra[3] - yb[3]));
    s1b += (fabsf(pa[0] - hb[0]) + fabsf(pa[1] - hb[1])) +
           (fabsf(pa[2] - hb[2]) + fabsf(pa[3] - hb[3]));
  }
  for (int i = (nv << 2) + tid; i < n; i += stride) {   // scalar tail (none for N=4M)
    s1a += fabsf(resnet[i] - yv[i]);
    s1b += fabsf(avgp[i] - avgh[i]);
  }

  // ---- Smoothness: segments of 6; two segments (48B) = 3 aligned b128 loads ----
  const float k6 = 1.0f / 6.0f;
  const int npair = nseg >> 1;
  const v4f* g4 = (const v4f*)gru;
  for (int p = tid; p < npair; p += stride) {
    const size_t base = 3 * (size_t)p;
    v4f x0 = NT(g4 + base + 0);
    v4f x1 = NT(g4 + base + 1);
    v4f x2 = NT(g4 + base + 2);
    float ma = ((x0[0] + x0[1]) + (x0[2] + x0[3]) + (x1[0] + x1[1])) * k6;
    float mb = ((x1[2] + x1[3]) + (x2[0] + x2[1]) + (x2[2] + x2[3])) * k6;
    s2 += (fabsf(x0[0] - ma) + fabsf(x0[1] - ma) + fabsf(x0[2] - ma) +
           fabsf(x0[3] - ma) + fabsf(x1[0] - ma) + fabsf(x1[1] - ma)) * k6;
    s2 += (fabsf(x1[2] - mb) + fabsf(x1[3] - mb) + fabsf(x2[0] - mb) +
           fabsf(x2[1] - mb) + fabsf(x2[2] - mb) + fabsf(x2[3] - mb)) * k6;
  }
  for (int t = (npair << 1) + tid; t < nseg; t += stride) {  // odd-segment tail
    const float* s = gru + 6 * (size_t)t;
    float a0 = s[0], a1 = s[1], a2 = s[2], a3 = s[3], a4 = s[4], a5 = s[5];
    float m = ((a0 + a1) + (a2 + a3) + (a4 + a5)) * k6;
    s2 += (fabsf(a0 - m) + fabsf(a1 - m) + fabsf(a2 - m) +
           fabsf(a3 - m) + fabsf(a4 - m) + fabsf(a5 - m)) * k6;
  }

  // ---- Wave reduction via fp32 WMMA (EXEC is all-ones here: no divergence) ----
  s1a = wave_sum32(s1a);
  s1b = wave_sum32(s1b);
  s2  = wave_sum32(s2);

  __shared__ float lds[3][32];
  const int lane = threadIdx.x & 31;
  const int wv   = threadIdx.x >> 5;
  if (lane == 0) { lds[0][wv] = s1a; lds[1][wv] = s1b; lds[2][wv] = s2; }
  __syncthreads();
  if (threadIdx.x == 0) {
    const int nw = blockDim.x >> 5;
    float t0 = 0.0f, t1 = 0.0f, t2 = 0.0f;
    for (int w = 0; w < nw; ++w) { t0 += lds[0][w]; t1 += lds[1][w]; t2 += lds[2][w]; }
    ws[0 * gridDim.x + blockIdx.x] = t0;   // SoA layout: coalesced in stage 2
    ws[1 * gridDim.x + blockIdx.x] = t1;
    ws[2 * gridDim.x + blockIdx.x] = t2;
  }
}

__global__ __launch_bounds__(256)
void tl_final(const float* __restrict__ ws, float* __restrict__ out,
              int nblocks, float invN) {
  float t0 = 0.0f, t1 = 0.0f, t2 = 0.0f;
  for (int i = threadIdx.x; i < nblocks; i += blockDim.x) {
    t0 += ws[0 * nblocks + i];
    t1 += ws[1 * nblocks + i];
    t2 += ws[2 * nblocks + i];
  }
  t0 = wave_sum32(t0);
  t1 = wave_sum32(t1);
  t2 = wave_sum32(t2);

  __shared__ float lds[3][32];
  const int lane = threadIdx.x & 31;
  const int wv   = threadIdx.x >> 5;
  if (lane == 0) { lds[0][wv] = t0; lds[1][wv] = t1; lds[2][wv] = t2; }
  __syncthreads();
  if (threadIdx.x == 0) {
    const int nw = blockDim.x >> 5;
    float s0 = 0.0f, s1 = 0.0f, s2 = 0.0f;
    for (int w = 0; w < nw; ++w) { s0 += lds[0][w]; s1 += lds[1][w]; s2 += lds[2][w]; }
    float l1     = (s0 + s1) * invN;   // mean(|r-y|) + mean(|ap-ah|), equal lengths
    float smooth = s2;
    out[0] = l1 + 100.0f * smooth;     // loss
    out[1] = l1;                       // l1_loss
    out[2] = smooth;                   // smooth_loss
  }
}

extern "C" void kernel_launch(void* const* d_in, const int* in_sizes, int n_in,
                              void* d_out, int out_size, void* d_ws, size_t ws_size,
                              hipStream_t stream) {
  const float* resnet = (const float*)d_in[0];
  const float* gru    = (const float*)d_in[1];
  const float* avgp   = (const float*)d_in[2];
  const float* yv     = (const float*)d_in[3];
  const float* avgh   = (const float*)d_in[4];
  float* ws  = (float*)d_ws;
  float* out = (float*)d_out;

  const int n    = in_sizes[0];
  const int nseg = in_sizes[1] / 6;

  // 1024 blocks x 256 threads = 262144 threads = 8192 wave32s: saturates the
  // device; each thread does 4 b128-quad L1 iterations + 8 segment-pair iters.
  int nblocks = 1024;
  const size_t need = (size_t)nblocks * 3u * sizeof(float);
  if (ws_size < need) {
    nblocks = (int)(ws_size / (3u * sizeof(float)));
    if (nblocks < 1) nblocks = 1;
  }

  tl_partial<<<dim3(nblocks), dim3(256), 0, stream>>>(resnet, gru, avgp, yv, avgh,
                                                      ws, n, nseg);
  tl_final<<<dim3(1), dim3(256), 0, stream>>>(ws, out, nblocks, 1.0f / (float)n);
}